// GCN2_9371618640574
// MI455X (gfx1250) — compile-verified
//
#include <hip/hip_runtime.h>

typedef __attribute__((ext_vector_type(2))) float v2f;
typedef __attribute__((ext_vector_type(8))) float v8f;

#define GN      100000
#define GE      1000000
#define FIN     128
#define HID     64
#define OUTC    40
#define NLAYERS 4
#define ALPHA_C 0.5f
#define BN_EPS  1e-5f

// ---------------------------------------------------------------- utilities
__global__ __launch_bounds__(256) void k_fill(float* __restrict__ p, int n, float v) {
  int i = blockIdx.x * 256 + threadIdx.x;
  if (i < n) p[i] = v;
}

__global__ __launch_bounds__(256) void k_deg(const int* __restrict__ col,
                                             float* __restrict__ deg, int e) {
  int i = blockIdx.x * 256 + threadIdx.x;
  if (i < e) atomicAdd(&deg[col[i]], 1.0f);
}

__global__ __launch_bounds__(256) void k_dinv(const float* __restrict__ deg,
                                              float* __restrict__ dinv, int n) {
  int i = blockIdx.x * 256 + threadIdx.x;
  if (i < n) {
    float d = deg[i];
    dinv[i] = (d > 0.f) ? rsqrtf(d) : 0.f;
  }
}

// ---------------------------------------------------------------- WMMA GEMM
// D = A[M x K] @ W[K x NC] (+ bias) (+ betaA * A, requires K == NC) (relu?)
// Compile-time K/NC/tiles: no EXEC-masked loads in the K-loop.
// One wave computes MT 16-row tiles x NT 16-col tiles; B fragments are
// shared across the MT row tiles. f32 WMMA 16x16x4:
//   A frag = {A[m][kb], A[m][kb+1]},  kb = k0 + 2*(lane>=16), m = r0+(lane&15)
//   B frag = {W[kb][n], W[kb+1][n]},  n  = 16*t + (lane&15)
//   C/D:  acc[t][i] = D[r0 + i + 8*(lane>=16)][16*t + (lane&15)]
template <int K, int NC, int NT, int MT>
__global__ __launch_bounds__(32)
void k_wmma_gemm(const float* __restrict__ A, const float* __restrict__ W,
                 const float* __restrict__ bias,
                 float* __restrict__ D0, float* __restrict__ D1,
                 float betaA, int doRelu) {
  const int r0   = blockIdx.x * (16 * MT);
  const int lane = threadIdx.x & 31;
  const int hi   = lane >> 4;  // 0 or 1
  const int ln   = lane & 15;

  v8f z = {};
  v8f acc[MT][NT];
#pragma unroll
  for (int r = 0; r < MT; ++r)
#pragma unroll
    for (int t = 0; t < NT; ++t) acc[r][t] = z;

  // Per-lane column index, clamped for the (NC % 16 != 0) tail tile so every
  // load is unconditional; invalid lanes contribute 0 via a VALU select.
  int  ncol[NT];
  bool nval[NT];
#pragma unroll
  for (int t = 0; t < NT; ++t) {
    const int n = t * 16 + ln;
    nval[t] = (n < NC);
    ncol[t] = nval[t] ? n : (NC - 1);
  }

  const float* arow[MT];
#pragma unroll
  for (int r = 0; r < MT; ++r)
    arow[r] = A + (size_t)(r0 + 16 * r + ln) * K + (hi << 1);

#pragma unroll
  for (int k0 = 0; k0 < K; k0 += 4) {
    const float* wrow = W + (size_t)(k0 + (hi << 1)) * NC;
    v2f b[NT];
#pragma unroll
    for (int t = 0; t < NT; ++t) {
      const float b0 = wrow[ncol[t]];
      const float b1 = wrow[NC + ncol[t]];
      b[t].x = nval[t] ? b0 : 0.f;
      b[t].y = nval[t] ? b1 : 0.f;
    }
#pragma unroll
    for (int r = 0; r < MT; ++r) {
      v2f a;
      a.x = arow[r][k0];
      a.y = arow[r][k0 + 1];
#pragma unroll
      for (int t = 0; t < NT; ++t)
        acc[r][t] = __builtin_amdgcn_wmma_f32_16x16x4_f32(
            false, a, false, b[t], (short)0, acc[r][t], false, false);
    }
  }

#pragma unroll
  for (int r = 0; r < MT; ++r) {
#pragma unroll
    for (int t = 0; t < NT; ++t) {
      const int n = t * 16 + ln;
      if (n < NC) {
        const float bv = bias ? bias[n] : 0.f;
#pragma unroll
        for (int i = 0; i < 8; ++i) {
          const int m = r0 + 16 * r + i + hi * 8;
          float v = acc[r][t][i] + bv;
          if (betaA != 0.f) v += betaA * A[(size_t)m * K + n];  // only when K==NC
          if (doRelu) v = fmaxf(v, 0.f);
          D0[(size_t)m * NC + n] = v;
          if (D1) D1[(size_t)m * NC + n] = v;
        }
      }
    }
  }
}

// ---------------------------------------------------------------- aggregation
// agg[i][:] += dinv[i]^2 * support[i][:]   (self-loop term, no atomics needed)
__global__ __launch_bounds__(256)
void k_selfloop(const float* __restrict__ dinv, const float* __restrict__ support,
                float* __restrict__ agg) {
  int idx = blockIdx.x * 256 + threadIdx.x;  // over GN*HID
  if (idx >= GN * HID) return;
  int i = idx >> 6;
  float dd = dinv[i];
  agg[idx] += dd * dd * support[idx];
}

// edge scatter: 16 threads per edge, float4 gather + 4 f32 atomics (L2-resident)
__global__ __launch_bounds__(256)
void k_scatter(const int* __restrict__ ei, const float* __restrict__ dinv,
               const float* __restrict__ support, float* __restrict__ agg) {
  int gid = blockIdx.x * 256 + threadIdx.x;
  int e = gid >> 4;
  if (e >= GE) return;
  int part = gid & 15;
  int r = ei[e];
  int c = ei[GE + e];
  float nrm = dinv[r] * dinv[c];
  const float4 v = *(const float4*)(support + (size_t)r * HID + part * 4);
  float* ap = agg + (size_t)c * HID + part * 4;
  atomicAdd(ap + 0, nrm * v.x);
  atomicAdd(ap + 1, nrm * v.y);
  atomicAdd(ap + 2, nrm * v.z);
  atomicAdd(ap + 3, nrm * v.w);
}

// ---------------------------------------------------------------- batch norm
__global__ __launch_bounds__(256)
void k_bn_stats(const float* __restrict__ agg, float* __restrict__ stats) {
  __shared__ float ls[256];
  __shared__ float ls2[256];
  const int ch = threadIdx.x & 63;
  const int rg = threadIdx.x >> 6;  // 0..3
  float s = 0.f, s2 = 0.f;
  for (int i = blockIdx.x * 4 + rg; i < GN; i += gridDim.x * 4) {
    float v = agg[(size_t)i * HID + ch];
    s += v;
    s2 += v * v;
  }
  ls[threadIdx.x] = s;
  ls2[threadIdx.x] = s2;
  __syncthreads();
  if (threadIdx.x < 64) {
    s  = ls[threadIdx.x] + ls[threadIdx.x + 64] + ls[threadIdx.x + 128] + ls[threadIdx.x + 192];
    s2 = ls2[threadIdx.x] + ls2[threadIdx.x + 64] + ls2[threadIdx.x + 128] + ls2[threadIdx.x + 192];
    atomicAdd(&stats[ch], s);
    atomicAdd(&stats[64 + ch], s2);
  }
}

__global__ __launch_bounds__(64)
void k_bn_finalize(const float* __restrict__ stats, const float* __restrict__ gamma,
                   const float* __restrict__ beta, float* __restrict__ coef) {
  int c = threadIdx.x;
  float mean = stats[c] * (1.0f / GN);
  float var  = stats[64 + c] * (1.0f / GN) - mean * mean;  // biased, matches jnp.var
  float sc   = gamma[c] * rsqrtf(var + BN_EPS);
  coef[c]      = sc;
  coef[64 + c] = beta[c] - mean * sc;
}

// hprev[i][f] += relu(scale[f]*agg[i][f] + shift[f])  (in-place residual)
__global__ __launch_bounds__(256)
void k_bn_apply(const float* __restrict__ agg, const float* __restrict__ coef,
                float* __restrict__ h) {
  int idx = blockIdx.x * 256 + threadIdx.x;
  if (idx >= GN * HID) return;
  int f = idx & 63;
  float v = agg[idx] * coef[f] + coef[64 + f];
  h[idx] += fmaxf(v, 0.f);
}

// ---------------------------------------------------------------- launcher
extern "C" void kernel_launch(void* const* d_in, const int* in_sizes, int n_in,
                              void* d_out, int out_size, void* d_ws, size_t ws_size,
                              hipStream_t stream) {
  const float* x     = (const float*)d_in[0];
  const int*   ei    = (const int*)d_in[1];   // [2, E] flat: rows at 0, cols at GE
  const float* Wi    = (const float*)d_in[2];
  const float* bi    = (const float*)d_in[3];
  const float* w1    = (const float*)d_in[4];  // [L,64,64]
  const float* w2    = (const float*)d_in[5];
  const float* gamma = (const float*)d_in[6];  // [L,64]
  const float* beta  = (const float*)d_in[7];
  const float* Wo    = (const float*)d_in[8];  // [64,40]
  const float* bo    = (const float*)d_in[9];
  float* out = (float*)d_out;

  float* ws      = (float*)d_ws;
  float* deg     = ws;
  float* dinv    = deg + GN;
  float* h0      = dinv + GN;
  float* hprev   = h0 + (size_t)GN * HID;
  float* support = hprev + (size_t)GN * HID;
  float* agg     = support + (size_t)GN * HID;
  float* stats   = agg + (size_t)GN * HID;  // 128
  float* coef    = stats + 128;             // 128

  const int NH = GN * HID;
  const int gemmGrid = GN / 32;  // MT=2 -> 32 rows per wave, 3125 blocks, exact

  // degree / normalization
  k_fill<<<(GN + 255) / 256, 256, 0, stream>>>(deg, GN, 1.0f);  // self-loops
  k_deg<<<(GE + 255) / 256, 256, 0, stream>>>(ei + GE, deg, GE);
  k_dinv<<<(GN + 255) / 256, 256, 0, stream>>>(deg, dinv, GN);

  // h = relu(x @ Wi + bi); h0 = h
  k_wmma_gemm<FIN, HID, 4, 2><<<gemmGrid, 32, 0, stream>>>(x, Wi, bi, hprev, h0,
                                                           0.f, 1);

  for (int l = 0; l < NLAYERS; ++l) {
    const float* W1l = w1 + (size_t)l * HID * HID;
    const float* W2l = w2 + (size_t)l * HID * HID;

    // support = hprev + hprev @ W1
    k_wmma_gemm<HID, HID, 4, 2><<<gemmGrid, 32, 0, stream>>>(
        hprev, W1l, nullptr, support, nullptr, 1.0f, 0);
    // agg = alpha*h0 + h0 @ W2   ("initial" term seeds the aggregation buffer)
    k_wmma_gemm<HID, HID, 4, 2><<<gemmGrid, 32, 0, stream>>>(
        h0, W2l, nullptr, agg, nullptr, ALPHA_C, 0);
    // self-loop contribution (deterministic, no atomics)
    k_selfloop<<<(NH + 255) / 256, 256, 0, stream>>>(dinv, support, agg);
    // edge messages: agg[col] += dinv[row]*dinv[col]*support[row]
    k_scatter<<<(GE * 16) / 256, 256, 0, stream>>>(ei, dinv, support, agg);

    // batch norm (training-mode batch stats) + relu + residual, in place
    k_fill<<<1, 256, 0, stream>>>(stats, 128, 0.f);
    k_bn_stats<<<512, 256, 0, stream>>>(agg, stats);
    k_bn_finalize<<<1, 64, 0, stream>>>(stats, gamma + l * HID, beta + l * HID, coef);
    k_bn_apply<<<(NH + 255) / 256, 256, 0, stream>>>(agg, coef, hprev);
  }

  // out = hprev @ Wo + bo   (NC=40 -> 3 col tiles, tail tile via clamped loads)
  k_wmma_gemm<HID, OUTC, 3, 2><<<gemmGrid, 32, 0, stream>>>(hprev, Wo, bo, out,
                                                            nullptr, 0.f, 0);
}